// KNNSelector_40681930227994
// MI455X (gfx1250) — compile-verified
//
#include <hip/hip_runtime.h>

// ---------------------------------------------------------------------------
// KNN retrieval for MI455X (gfx1250, wave32).
//   d2[q,n] = |q|^2 - 2 q.k + |k|^2 ; ranking only needs s[n] = |k|^2 - 2 q.k
//   -> f32 WMMA (V_WMMA_F32_16X16X4_F32) for the 256x50000x64 dot products,
//      block-local top-8, cross-chunk merge, float4 gather of obs rows.
// ---------------------------------------------------------------------------

typedef float v2f __attribute__((ext_vector_type(2)));
typedef float v8f __attribute__((ext_vector_type(8)));

#define NQ      256
#define DIM     64
#define NKEYS   50000
#define KRET    8
#define OBSF    3072                     // 3*32*32 floats per observation
#define CHUNK   512                      // keys per block in score kernel
#define NCHUNKS ((NKEYS + CHUNK - 1) / CHUNK)   // 98
#define QT      16                       // queries per block
#define BIGF    3.4e38f

// ---------------------------------------------------------------- kernel 0 --
__global__ __launch_bounds__(256) void ksq_kernel(const float* __restrict__ k,
                                                  float* __restrict__ ksq) {
  int n = blockIdx.x * blockDim.x + threadIdx.x;
  if (n >= NKEYS) return;
  const float4* row = reinterpret_cast<const float4*>(k + (size_t)n * DIM);
  float s = 0.f;
#pragma unroll
  for (int i = 0; i < DIM / 4; ++i) {
    float4 v = row[i];
    s += v.x * v.x + v.y * v.y + v.z * v.z + v.w * v.w;
  }
  ksq[n] = s;
}

// ---------------------------------------------------------------- kernel 1 --
// grid = (NCHUNKS, NQ/QT), block = 256 (8 waves).
// Each wave owns 64 keys (4 tiles of 16); each block owns 16 queries.
__global__ __launch_bounds__(256) void score_topk_kernel(
    const float* __restrict__ q, const float* __restrict__ k,
    const float* __restrict__ ksq,
    float* __restrict__ candS, int* __restrict__ candI) {
  __shared__ float s_score[QT * CHUNK];          // 32 KB
  __shared__ float s_cs[QT * 16 * KRET];         // 8 KB
  __shared__ int   s_ci[QT * 16 * KRET];         // 8 KB

  const int lane = threadIdx.x & 31;
  const int wave = threadIdx.x >> 5;
  const int h    = lane >> 4;                    // lane half (0/1)
  const int m    = lane & 15;                    // A-row / B-col within tile
  const int q0   = blockIdx.y * QT;
  const int n0   = blockIdx.x * CHUNK;

  // Hoist A fragments: aq[kk] covers K-dims [4kk, 4kk+4).
  // ISA 7.12.2 (32-bit A 16x4): a[v] = Q[m][kbase + 2h + v].
  v2f aq[16];
#pragma unroll
  for (int kk = 0; kk < 16; ++kk) {
    const float* qp = q + (q0 + m) * DIM + kk * 4 + 2 * h;
    v2f a; a.x = qp[0]; a.y = qp[1];
    aq[kk] = a;
  }

#pragma unroll
  for (int kt = 0; kt < 4; ++kt) {               // 4 key tiles per wave
    const int nbase = n0 + wave * 64 + kt * 16;
    const int ng    = nbase + m;                 // this lane's key (B column)
    const bool valid = (ng < NKEYS);
    const float* kp = k + (size_t)(valid ? ng : 0) * DIM + 2 * h;

    v8f acc = {};
#if defined(__has_builtin) && __has_builtin(__builtin_amdgcn_wmma_f32_16x16x4_f32)
#pragma unroll
    for (int kk = 0; kk < 16; ++kk) {            // K = 64 in steps of 4
      v2f b; b.x = kp[kk * 4]; b.y = kp[kk * 4 + 1];
      acc = __builtin_amdgcn_wmma_f32_16x16x4_f32(
          /*neg_a=*/false, aq[kk], /*neg_b=*/false, b,
          /*c_mod=*/(short)0, acc, /*reuse_a=*/false, /*reuse_b=*/false);
    }
#else
    // Fallback (compile-probe only): keeps VGPR shape, not a real 16x16 GEMM.
#pragma unroll
    for (int kk = 0; kk < 16; ++kk) {
      v2f b; b.x = kp[kk * 4]; b.y = kp[kk * 4 + 1];
#pragma unroll
      for (int r = 0; r < 8; ++r) acc[r] += aq[kk].x * b.x + aq[kk].y * b.y;
    }
#endif

    // Epilogue: score = |k|^2 - 2*dot.  D layout: VGPR r -> row (r + 8h),
    // column = lane&15.  Per-lane validity via selects (EXEC stays all-ones).
    const float kn = valid ? ksq[ng] : BIGF;
    const int   ln = wave * 64 + kt * 16 + m;    // local column in [0, CHUNK)
#pragma unroll
    for (int r = 0; r < 8; ++r) {
      const float sc = valid ? (kn - 2.0f * acc[r]) : BIGF;
      s_score[(r + 8 * h) * CHUNK + ln] = sc;
    }
  }
  __syncthreads();

  // Stage 1: 16 scanner-threads per query, each scans CHUNK/16 = 32 scores.
  {
    const int qsel = threadIdx.x >> 4;           // 0..15
    const int sub  = threadIdx.x & 15;           // 0..15
    float bs[KRET]; int bi[KRET];
#pragma unroll
    for (int i = 0; i < KRET; ++i) { bs[i] = BIGF; bi[i] = 0; }
    const float* row = s_score + qsel * CHUNK;
    for (int j = sub * (CHUNK / 16); j < (sub + 1) * (CHUNK / 16); ++j) {
      const float s = row[j];
      if (s < bs[KRET - 1]) {
        int p = KRET - 1;
        while (p > 0 && bs[p - 1] > s) { bs[p] = bs[p - 1]; bi[p] = bi[p - 1]; --p; }
        bs[p] = s; bi[p] = n0 + j;
      }
    }
#pragma unroll
    for (int i = 0; i < KRET; ++i) {
      s_cs[(qsel * 16 + sub) * KRET + i] = bs[i];
      s_ci[(qsel * 16 + sub) * KRET + i] = bi[i];
    }
  }
  __syncthreads();

  // Stage 2: leader per query merges 16*8 = 128 candidates -> chunk top-8.
  if ((threadIdx.x & 15) == 0) {
    const int qsel = threadIdx.x >> 4;
    float bs[KRET]; int bi[KRET];
#pragma unroll
    for (int i = 0; i < KRET; ++i) { bs[i] = BIGF; bi[i] = 0; }
    for (int c = 0; c < 16 * KRET; ++c) {
      const float s = s_cs[qsel * 16 * KRET + c];
      const int  ix = s_ci[qsel * 16 * KRET + c];
      if (s < bs[KRET - 1]) {
        int p = KRET - 1;
        while (p > 0 && bs[p - 1] > s) { bs[p] = bs[p - 1]; bi[p] = bi[p - 1]; --p; }
        bs[p] = s; bi[p] = ix;
      }
    }
    const size_t base = ((size_t)(q0 + qsel) * NCHUNKS + blockIdx.x) * KRET;
#pragma unroll
    for (int i = 0; i < KRET; ++i) {
      candS[base + i] = bs[i];
      candI[base + i] = bi[i];
    }
  }
}

// ---------------------------------------------------------------- kernel 2 --
// One thread per query: merge NCHUNKS*8 candidates -> final top-8 (ascending).
__global__ __launch_bounds__(256) void merge_kernel(
    const float* __restrict__ candS, const int* __restrict__ candI,
    int* __restrict__ finalI) {
  const int qi = blockIdx.x * blockDim.x + threadIdx.x;
  if (qi >= NQ) return;
  float bs[KRET]; int bi[KRET];
#pragma unroll
  for (int i = 0; i < KRET; ++i) { bs[i] = BIGF; bi[i] = 0; }
  const size_t base = (size_t)qi * NCHUNKS * KRET;
  for (int c = 0; c < NCHUNKS * KRET; ++c) {
    const float s = candS[base + c];
    if (s < bs[KRET - 1]) {
      const int ix = candI[base + c];
      int p = KRET - 1;
      while (p > 0 && bs[p - 1] > s) { bs[p] = bs[p - 1]; bi[p] = bi[p - 1]; --p; }
      bs[p] = s; bi[p] = ix;
    }
  }
#pragma unroll
  for (int i = 0; i < KRET; ++i) finalI[qi * KRET + i] = bi[i];
}

// ---------------------------------------------------------------- kernel 3 --
// out[j][q][:] = obs[idx[q][j]][:]  -- the bandwidth-dominant 25+25 MB move.
__global__ __launch_bounds__(256) void gather_kernel(
    const float* __restrict__ obs, const int* __restrict__ finalI,
    float* __restrict__ out) {
  const int b  = blockIdx.x;           // 0..2047 = (j, q)
  const int j  = b >> 8;
  const int qi = b & 255;
  int idx = finalI[qi * KRET + j];
  if (idx < 0 || idx >= NKEYS) idx = 0;
  const float4* src = reinterpret_cast<const float4*>(obs) + (size_t)idx * (OBSF / 4);
  float4* dst = reinterpret_cast<float4*>(out) + ((size_t)j * NQ + qi) * (OBSF / 4);
  __builtin_prefetch(src + threadIdx.x, 0, 0);   // global_prefetch_b8
#pragma unroll
  for (int i = 0; i < OBSF / 4 / 256; ++i)       // 3 float4 per thread
    dst[threadIdx.x + 256 * i] = src[threadIdx.x + 256 * i];
}

// ------------------------------------------------------------------ launch --
extern "C" void kernel_launch(void* const* d_in, const int* in_sizes, int n_in,
                              void* d_out, int out_size, void* d_ws, size_t ws_size,
                              hipStream_t stream) {
  const float* q   = (const float*)d_in[0];   // [256, 64]
  const float* k   = (const float*)d_in[1];   // [50000, 64]
  const float* obs = (const float*)d_in[2];   // [50000, 3, 32, 32]
  float* out = (float*)d_out;                 // [8, 256, 3, 32, 32]

  // Workspace layout (bytes):
  //   [0, 200704)           ksq        50000 f32 (padded)
  //   [200704, 1003520)     candS      256*98*8 f32
  //   [1003520, 1806336)    candI      256*98*8 i32
  //   [1806336, 1814528)    finalI     256*8    i32
  char* ws = (char*)d_ws;
  float* ksq    = (float*)(ws);
  float* candS  = (float*)(ws + 200704);
  int*   candI  = (int*)  (ws + 1003520);
  int*   finalI = (int*)  (ws + 1806336);

  ksq_kernel<<<(NKEYS + 255) / 256, 256, 0, stream>>>(k, ksq);

  dim3 grid1(NCHUNKS, NQ / QT);
  score_topk_kernel<<<grid1, 256, 0, stream>>>(q, k, ksq, candS, candI);

  merge_kernel<<<1, 256, 0, stream>>>(candS, candI, finalI);

  gather_kernel<<<NQ * KRET, 256, 0, stream>>>(obs, finalI, out);
}